// DecayFlow_49727131353203
// MI455X (gfx1250) — compile-verified
//
#include <hip/hip_runtime.h>

typedef __attribute__((ext_vector_type(2))) float v2f;
typedef __attribute__((ext_vector_type(8))) float v8f;
typedef __attribute__((ext_vector_type(4))) unsigned int v4u;
typedef __attribute__((ext_vector_type(8))) unsigned int v8u;

static constexpr int kD       = 64;       // state dim
static constexpr int kQBits   = 8;        // dt = q/256 + r, q in [0,255]
static constexpr int kTayMain = 12;       // Taylor order for expm(-A/256)
static constexpr int kTayRem  = 8;        // Taylor order for expm(-r*A), r < 1/256
static constexpr float kDelta = 1.0f / 256.0f;

__device__ __forceinline__ v8f wmma_f32_16x16x4(v2f a, v2f b, v8f c) {
  // 8 args: (neg_a, A, neg_b, B, c_mod, C, reuse_a, reuse_b)
  return __builtin_amdgcn_wmma_f32_16x16x4_f32(false, a, false, b, (short)0, c,
                                               false, false);
}

// ---------------------------------------------------------------------------
// TDM: DMA one contiguous 64x64 f32 matrix (16 KB) from global into LDS.
// D# built per CDNA5 ISA section 8.3/8.4:
//   group0: count=1 | lds_addr | global_addr[56:0] | type=2
//   group1: workgroup_mask=0, data_size=2 (4B), tensor 64x64, tile 64x64,
//           dim0 stride 64 (contiguous rows)
// Issued once per wave (EXEC ignored); tracked with TENSORcnt.
// ---------------------------------------------------------------------------
__device__ __forceinline__ void tdm_load_mat64(const float* gsrc,
                                               unsigned int lds_byte_off) {
  const unsigned long long ga = (unsigned long long)(uintptr_t)gsrc;
  v4u g0;
  g0.x = 0x1u;                               // count=1, user descriptor
  g0.y = lds_byte_off;                       // lds_addr (bytes)
  g0.z = (unsigned int)(ga & 0xffffffffu);   // global_addr[31:0]
  g0.w = (unsigned int)(ga >> 32) | (2u << 30);  // global_addr[56:32] | type=2
  v8u g1;
  g1.s0 = 0x00020000u;      // workgroup_mask=0 | data_size=2 (4B) | no flags
  g1.s1 = 64u << 16;        // barrier_addr=0 | tensor_dim0 = 64
  g1.s2 = 64u << 16;        // tensor_dim0_hi=0 | tensor_dim1 = 64
  g1.s3 = 64u << 16;        // tensor_dim1_hi=0 | tile_dim0 = 64
  g1.s4 = 64u;              // tile_dim1 = 64 | tile_dim2 = 0
  g1.s5 = 64u;              // tensor_dim0_stride = 64 (contiguous rows)
  g1.s6 = 4096u << 16;      // stride0_hi=0 | tensor_dim1_stride = 4096
  g1.s7 = 0u;               // stride1_hi = 0
  asm volatile("tensor_load_to_lds %0, %1" ::"s"(g0), "s"(g1) : "memory");
}

// ---------------------------------------------------------------------------
// Phase 0 (one workgroup, 16 waves): build E_j = expm(-2^j * delta * A),
// j = 0..7, into workspace. Horner Taylor for E_0, then repeated squaring.
// Each wave owns one 16x16 output tile of the 64x64 matmuls.
// ---------------------------------------------------------------------------
__global__ __launch_bounds__(512) void DecayFlow_build_tables(
    const float* __restrict__ A, float* __restrict__ E) {
  __shared__ float sT[kD * kD];  // -delta * A
  __shared__ float sS[kD * kD];  // running matrix
  __shared__ float sP[kD * kD];  // product scratch
  const int tid  = threadIdx.x;
  const int lane = tid & 31;
  const int wv   = tid >> 5;        // 0..15
  const int ti   = wv >> 2;         // output row tile
  const int tj   = wv & 3;          // output col tile
  const int n    = lane & 15;
  const int hi   = lane >> 4;

  for (int e = tid; e < kD * kD; e += 512) {
    sT[e] = -kDelta * A[e];
    sS[e] = ((e >> 6) == (e & 63)) ? 1.0f : 0.0f;  // identity
  }
  __syncthreads();

  // S = I + X(I + X/2(I + X/3(...))),  X = -delta*A  (order kTayMain)
  for (int k = kTayMain; k >= 1; --k) {
    v8f c = {};
#pragma unroll
    for (int kk = 0; kk < 16; ++kk) {
      const int k0 = kk * 4 + 2 * hi;
      v2f a, b;
      a.x = sT[(ti * 16 + n) * kD + k0];
      a.y = sT[(ti * 16 + n) * kD + k0 + 1];
      b.x = sS[(k0)     * kD + tj * 16 + n];
      b.y = sS[(k0 + 1) * kD + tj * 16 + n];
      c = wmma_f32_16x16x4(a, b, c);
    }
#pragma unroll
    for (int v = 0; v < 8; ++v)
      sP[(ti * 16 + v + 8 * hi) * kD + tj * 16 + n] = c[v];
    __syncthreads();
    const float inv = 1.0f / (float)k;
    for (int e = tid; e < kD * kD; e += 512)
      sS[e] = (((e >> 6) == (e & 63)) ? 1.0f : 0.0f) + sP[e] * inv;
    __syncthreads();
  }

  for (int e = tid; e < kD * kD; e += 512) E[e] = sS[e];  // E_0

  for (int j = 1; j < kQBits; ++j) {
    v8f c = {};
#pragma unroll
    for (int kk = 0; kk < 16; ++kk) {
      const int k0 = kk * 4 + 2 * hi;
      v2f a, b;
      a.x = sS[(ti * 16 + n) * kD + k0];
      a.y = sS[(ti * 16 + n) * kD + k0 + 1];
      b.x = sS[(k0)     * kD + tj * 16 + n];
      b.y = sS[(k0 + 1) * kD + tj * 16 + n];
      c = wmma_f32_16x16x4(a, b, c);
    }
#pragma unroll
    for (int v = 0; v < 8; ++v)
      sP[(ti * 16 + v + 8 * hi) * kD + tj * 16 + n] = c[v];
    __syncthreads();
    for (int e = tid; e < kD * kD; e += 512) sS[e] = sP[e];
    __syncthreads();
    for (int e = tid; e < kD * kD; e += 512) E[j * kD * kD + e] = sS[e];
  }
}

// 64x16 <- 64x64 @ 64x16 via sixteen K=4 WMMA steps per 16-row output chunk;
// D-frags staged to this wave's LDS slab in plain [k][n] layout.
__device__ __forceinline__ void wave_matvec(const float* __restrict__ sM,
                                            const v2f* xf, float* myY,
                                            int n, int hi) {
  for (int i = 0; i < 4; ++i) {
    v8f c = {};
#pragma unroll
    for (int kk = 0; kk < 16; ++kk) {
      const int k0 = 4 * kk + 2 * hi;
      v2f a;
      a.x = sM[(i * 16 + n) * kD + k0];
      a.y = sM[(i * 16 + n) * kD + k0 + 1];
      c = wmma_f32_16x16x4(a, xf[kk], c);
    }
#pragma unroll
    for (int v = 0; v < 8; ++v)
      myY[(i * 16 + v + 8 * hi) * 16 + n] = c[v];
  }
}

// ---------------------------------------------------------------------------
// Phase 1: each wave carries 16 samples as a 64x16 column block (B operand,
// sixteen v2f B-fragments). Apply expm(-r*A) via Taylor, then the E_j chain
// with per-sample (per-lane) blending. Stage matrices arrive via TDM
// (tensor_load_to_lds) into a double-buffered LDS slot, prefetched one stage
// ahead so the DMA runs under the WMMA compute.
// Dynamic-LDS layout (base offset 0): [0..4095] buf0, [4096..8191] buf1,
// [8192 + wv*1024 ...] per-wave 64x16 staging.
// ---------------------------------------------------------------------------
__global__ __launch_bounds__(128) void DecayFlow_apply(
    const float* __restrict__ hidden, const float* __restrict__ dt,
    const float* __restrict__ E, const float* __restrict__ A,
    float* __restrict__ out) {
  extern __shared__ float smem[];
  const int tid  = threadIdx.x;
  const int lane = tid & 31;
  const int wv   = tid >> 5;          // 0..3
  const int n    = lane & 15;
  const int hi   = lane >> 4;
  const int s    = (blockIdx.x * 4 + wv) * 16 + n;  // this lane's sample

  float* myY = smem + 2 * kD * kD + wv * (16 * kD);

  // Kick off DMA of A (-> buf0) and E_0 (-> buf1); they complete in order.
  if (wv == 0) {
    tdm_load_mat64(A, 0u);
    tdm_load_mat64(E, (unsigned int)(kD * kD * sizeof(float)));
  }

  // Overlap: load state vectors + per-sample scalars while the DMA runs.
  v2f x[16];
#pragma unroll
  for (int kk = 0; kk < 16; ++kk) {
    const int k0 = 4 * kk + 2 * hi;
    x[kk].x = hidden[s * kD + k0];
    x[kk].y = hidden[s * kD + k0 + 1];
  }
  const float dtv = dt[s];
  int q = (int)(dtv * 256.0f);
  q = q < 0 ? 0 : (q > 255 ? 255 : q);
  const float r = dtv - (float)q * kDelta;  // remainder, 0 <= r < 1/256

  if (wv == 0) __builtin_amdgcn_s_wait_tensorcnt(1);  // A landed (E_0 may fly)
  __syncthreads();

  // ---- Taylor remainder: x <- expm(-r*A) x  (||r*A|| <= 1), buf0 holds A ---
  constexpr float kInvK[kTayRem + 1] = {0.0f,        1.0f, 0.5f,
                                        1.0f / 3.0f, 0.25f, 0.2f,
                                        1.0f / 6.0f, 1.0f / 7.0f, 0.125f};
  v2f acc[16], t[16];
#pragma unroll
  for (int kk = 0; kk < 16; ++kk) { acc[kk] = x[kk]; t[kk] = x[kk]; }
#pragma unroll
  for (int k = 1; k <= kTayRem; ++k) {
    wave_matvec(smem, t, myY, n, hi);
    const float sc = -r * kInvK[k];  // constant 1/k: no divide in the loop
#pragma unroll
    for (int kk = 0; kk < 16; ++kk) {
      const int k0 = 4 * kk + 2 * hi;
      t[kk].x = sc * myY[k0 * 16 + n];
      t[kk].y = sc * myY[(k0 + 1) * 16 + n];
      acc[kk].x += t[kk].x;
      acc[kk].y += t[kk].y;
    }
  }
#pragma unroll
  for (int kk = 0; kk < 16; ++kk) x[kk] = acc[kk];

  if (wv == 0) __builtin_amdgcn_s_wait_tensorcnt(0);  // E_0 landed in buf1
  __syncthreads();

  // ---- Binary chain: stage j reads buf[(j+1)&1]; prefetch E_{j+1} into
  //      buf[j&1] (its last readers finished before the barrier we just hit).
  for (int j = 0; j < kQBits; ++j) {
    const float* cur = smem + ((j + 1) & 1) * (kD * kD);
    if (wv == 0 && j + 1 < kQBits)
      tdm_load_mat64(E + (j + 1) * kD * kD,
                     (unsigned int)((j & 1) * kD * kD * sizeof(float)));

    wave_matvec(cur, x, myY, n, hi);

    const bool bs = (q >> j) & 1;  // per-lane select -> v_cndmask, no diverge
#pragma unroll
    for (int kk = 0; kk < 16; ++kk) {
      const int k0 = 4 * kk + 2 * hi;
      const float yx = myY[k0 * 16 + n];
      const float yy = myY[(k0 + 1) * 16 + n];
      x[kk].x = bs ? yx : x[kk].x;
      x[kk].y = bs ? yy : x[kk].y;
    }

    if (wv == 0) __builtin_amdgcn_s_wait_tensorcnt(0);
    __syncthreads();
  }

  // Store result (each element written exactly once across the wave).
#pragma unroll
  for (int kk = 0; kk < 16; ++kk) {
    const int k0 = 4 * kk + 2 * hi;
    out[s * kD + k0]     = x[kk].x;
    out[s * kD + k0 + 1] = x[kk].y;
  }
}

extern "C" void kernel_launch(void* const* d_in, const int* in_sizes, int n_in,
                              void* d_out, int out_size, void* d_ws, size_t ws_size,
                              hipStream_t stream) {
  const float* hidden = (const float*)d_in[0];  // [B,64] f32
  const float* dt     = (const float*)d_in[1];  // [B]    f32
  const float* A      = (const float*)d_in[2];  // [64,64] f32
  float* out = (float*)d_out;
  float* E   = (float*)d_ws;                    // 8 * 64*64 f32 = 128 KB

  const int B = in_sizes[1];          // 8192
  const int blocks = B / 64;          // 4 waves/block * 16 samples/wave

  // Dynamic LDS: 2 stage buffers (2*16KB) + 4 per-wave 64x16 slabs (16KB)
  const size_t lds_bytes = (2 * kD * kD + 4 * 16 * kD) * sizeof(float);

  DecayFlow_build_tables<<<1, 512, 0, stream>>>(A, E);
  DecayFlow_apply<<<blocks, 128, lds_bytes, stream>>>(hidden, dt, E, A, out);
}